// GCN_87462714015857
// MI455X (gfx1250) — compile-verified
//
#include <hip/hip_runtime.h>

typedef float v2f __attribute__((ext_vector_type(2)));
typedef float v8f __attribute__((ext_vector_type(8)));

#define N_NODES 20000
#define N_EDGES 320000
#define XDIM    128
#define H       256
#define LNUM    4
#define GNUM    128
#define C_OUT   10
#define EPSV    1e-5f

// ---------------- utility fills ----------------
__global__ void k_fill(float* __restrict__ p, float v, int n) {
    int i = blockIdx.x * blockDim.x + threadIdx.x;
    if (i < n) p[i] = v;
}

// ---------------- degree / norm ----------------
__global__ void k_deg(const int* __restrict__ ei, float* __restrict__ deg) {
    int e = blockIdx.x * blockDim.x + threadIdx.x;
    if (e < N_EDGES) atomicAdd(&deg[ei[N_EDGES + e]], 1.0f);
}

__global__ void k_dinv(float* __restrict__ deg) {
    int i = blockIdx.x * blockDim.x + threadIdx.x;
    if (i < N_NODES) deg[i] = rsqrtf(deg[i]);
}

// ---------------- WMMA fp32 GEMM: C[MxH] = A[MxK] @ W[KxH] (+bias) ------------
// Requirements: M % 32 == 0, K % 4 == 0, output width == H (256).
// Block = 256 threads = 8 waves; each wave computes a 16x64 tile.
__global__ void __launch_bounds__(256) k_gemm256(
    const float* __restrict__ A, const float* __restrict__ W,
    const float* __restrict__ bias, float* __restrict__ Cc, int K) {
    const int lane = threadIdx.x & 31;
    const int wave = threadIdx.x >> 5;
    const int lr   = lane & 15;     // row/col within half
    const int half = lane >> 4;     // 0: K=+0/+1, 1: K=+2/+3
    const int m0   = blockIdx.x * 32 + (wave >> 2) * 16;
    const int n0   = (wave & 3) * 64;

    v8f acc[4] = {};
    const float* arow = A + (size_t)(m0 + lr) * K + 2 * half;
    for (int k0 = 0; k0 < K; k0 += 4) {
        // A fragment: 16x4 f32, row m0+lr, contiguous pair of K values
        v2f a = *(const v2f*)(arow + k0);
        // B fragments: 4x16 f32 per 16-wide column slice
        const float* w0 = W + (size_t)(k0 + 2 * half) * H + n0 + lr;
        const float* w1 = w0 + H;
#pragma unroll
        for (int nt = 0; nt < 4; ++nt) {
            v2f b = { w0[16 * nt], w1[16 * nt] };
            acc[nt] = __builtin_amdgcn_wmma_f32_16x16x4_f32(
                false, a, false, b, (short)0, acc[nt], false, false);
        }
    }
#pragma unroll
    for (int nt = 0; nt < 4; ++nt) {
        int col = n0 + nt * 16 + lr;
        float bv = bias ? bias[col] : 0.0f;
#pragma unroll
        for (int r = 0; r < 8; ++r) {
            int row = m0 + r + 8 * half;   // D layout: VGPR r -> M=r (lanes 0-15), M=r+8 (16-31)
            Cc[(size_t)row * H + col] = acc[nt][r] + bv;
        }
    }
}

// ---------------- agg = t * self_norm + convB ----------------
__global__ void k_self_init(const float* __restrict__ t, const float* __restrict__ dinv,
                            const float* __restrict__ convB, float* __restrict__ agg) {
    int i = blockIdx.x * blockDim.x + threadIdx.x;
    if (i >= N_NODES * H) return;
    int n = i / H, c = i % H;
    float dv = dinv[n];
    agg[i] = t[i] * dv * dv + convB[c];
}

// ---------------- edge scatter-add: one wave per edge ----------------
__global__ void k_edge_agg(const int* __restrict__ ei, const float* __restrict__ dinv,
                           const float* __restrict__ t, float* __restrict__ agg) {
    int gid  = blockIdx.x * blockDim.x + threadIdx.x;
    int e    = gid >> 5;
    int lane = gid & 31;
    if (e >= N_EDGES) return;
    int s = ei[e];
    int d = ei[N_EDGES + e];
    float nrm = dinv[s] * dinv[d];
    const float* trow = t + (size_t)s * H;
    float* arow = agg + (size_t)d * H;
#pragma unroll
    for (int j = 0; j < 8; ++j) {
        int c = lane + 32 * j;
        atomicAdd(&arow[c], trow[c] * nrm);
    }
}

// ---------------- BN over N rows: column partial sums ----------------
__global__ void k_bn_reduce(const float* __restrict__ x,
                            float* __restrict__ colsum, float* __restrict__ colsq) {
    int c  = threadIdx.x;            // 256 threads = 1 per column
    int r0 = blockIdx.x * 64;
    int r1 = min(r0 + 64, N_NODES);
    float s = 0.0f, q = 0.0f;
    for (int r = r0; r < r1; ++r) {
        float v = x[(size_t)r * H + c];
        s += v; q += v * v;
    }
    atomicAdd(&colsum[c], s);
    atomicAdd(&colsq[c], q);
}

__global__ void k_bn_apply(const float* __restrict__ x,
                           const float* __restrict__ colsum, const float* __restrict__ colsq,
                           const float* __restrict__ gamma, const float* __restrict__ beta,
                           float* __restrict__ out) {
    int i = blockIdx.x * blockDim.x + threadIdx.x;
    if (i >= N_NODES * H) return;
    int c = i % H;
    const float invN = 1.0f / (float)N_NODES;
    float mean = colsum[c] * invN;
    float var  = colsq[c] * invN - mean * mean;
    float v = gamma[c] * (x[i] - mean) * rsqrtf(var + EPSV) + beta[c];
    out[i] = v > 0.0f ? v : 0.0f;
}

// ---------------- global mean pool ----------------
__global__ void k_cnt(const int* __restrict__ batch, float* __restrict__ gcnt) {
    int n = blockIdx.x * blockDim.x + threadIdx.x;
    if (n < N_NODES) atomicAdd(&gcnt[batch[n]], 1.0f);
}

__global__ void k_pool_accum(const float* __restrict__ h, const int* __restrict__ batch,
                             float* __restrict__ gsum) {
    int i = blockIdx.x * blockDim.x + threadIdx.x;
    if (i >= N_NODES * H) return;
    int n = i / H, c = i % H;
    atomicAdd(&gsum[(size_t)batch[n] * H + c], h[i]);
}

__global__ void k_pool_div(float* __restrict__ gsum, const float* __restrict__ gcnt) {
    int i = blockIdx.x * blockDim.x + threadIdx.x;
    if (i >= GNUM * H) return;
    gsum[i] = gsum[i] / fmaxf(gcnt[i / H], 1.0f);
}

// ---------------- BN over G=128 rows (one block, 1 thread per column) --------
__global__ void k_bn_small(const float* __restrict__ x, const float* __restrict__ gamma,
                           const float* __restrict__ beta, float* __restrict__ out, int relu) {
    int c = threadIdx.x;             // H threads
    float s = 0.0f, q = 0.0f;
    for (int g = 0; g < GNUM; ++g) {
        float v = x[(size_t)g * H + c];
        s += v; q += v * v;
    }
    float mean = s * (1.0f / GNUM);
    float var  = q * (1.0f / GNUM) - mean * mean;
    float inv  = rsqrtf(var + EPSV);
    float gm = gamma[c], bt = beta[c];
    for (int g = 0; g < GNUM; ++g) {
        float v = gm * (x[(size_t)g * H + c] - mean) * inv + bt;
        if (relu) v = v > 0.0f ? v : 0.0f;
        out[(size_t)g * H + c] = v;
    }
}

// ---------------- final [128x256] @ [256x10] + bias ----------------
__global__ void k_fc3(const float* __restrict__ m, const float* __restrict__ W3,
                      const float* __restrict__ b3, float* __restrict__ out) {
    int i = blockIdx.x * blockDim.x + threadIdx.x;
    if (i >= GNUM * C_OUT) return;
    int g = i / C_OUT, c = i % C_OUT;
    float s = b3[c];
    for (int k = 0; k < H; ++k) s += m[(size_t)g * H + k] * W3[k * C_OUT + c];
    out[i] = s;
}

static inline int nblk(long long n, int b) { return (int)((n + b - 1) / b); }

extern "C" void kernel_launch(void* const* d_in, const int* in_sizes, int n_in,
                              void* d_out, int out_size, void* d_ws, size_t ws_size,
                              hipStream_t stream) {
    const float* x     = (const float*)d_in[0];
    const int*   ei    = (const int*)d_in[1];
    const int*   batch = (const int*)d_in[2];
    const float* encW  = (const float*)d_in[3];
    const float* encB  = (const float*)d_in[4];
    const float* convW = (const float*)d_in[5];
    const float* convB = (const float*)d_in[6];
    const float* bnG   = (const float*)d_in[7];
    const float* bnB   = (const float*)d_in[8];
    const float* fcW1  = (const float*)d_in[9];
    const float* fcB1  = (const float*)d_in[10];
    const float* fcG1  = (const float*)d_in[11];
    const float* fcBe1 = (const float*)d_in[12];
    const float* fcW2  = (const float*)d_in[13];
    const float* fcB2  = (const float*)d_in[14];
    const float* fcG2  = (const float*)d_in[15];
    const float* fcBe2 = (const float*)d_in[16];
    const float* fcW3  = (const float*)d_in[17];
    const float* fcB3  = (const float*)d_in[18];

    float* ws     = (float*)d_ws;
    float* deg    = ws;                                   // N
    float* h      = deg + N_NODES;                        // N*H
    float* t      = h + (size_t)N_NODES * H;              // N*H
    float* agg    = t + (size_t)N_NODES * H;              // N*H
    float* colsum = agg + (size_t)N_NODES * H;            // H
    float* colsq  = colsum + H;                           // H
    float* gsum   = colsq + H;                            // G*H
    float* gcnt   = gsum + (size_t)GNUM * H;              // G
    float* m1     = gcnt + GNUM;                          // G*H
    float* m2     = m1 + (size_t)GNUM * H;                // G*H

    // degree + symmetric-norm factor
    k_fill<<<nblk(N_NODES, 256), 256, 0, stream>>>(deg, 1.0f, N_NODES);
    k_deg <<<nblk(N_EDGES, 256), 256, 0, stream>>>(ei, deg);
    k_dinv<<<nblk(N_NODES, 256), 256, 0, stream>>>(deg);

    // encoder: h = x @ encW + encB
    k_gemm256<<<N_NODES / 32, 256, 0, stream>>>(x, encW, encB, h, XDIM);

    for (int i = 0; i < LNUM; ++i) {
        // t = h @ convW[i]
        k_gemm256<<<N_NODES / 32, 256, 0, stream>>>(h, convW + (size_t)i * H * H, nullptr, t, H);
        // agg = t * self_norm + convB[i]
        k_self_init<<<nblk((long long)N_NODES * H, 256), 256, 0, stream>>>(t, deg, convB + i * H, agg);
        // agg += scatter(t[src] * enorm -> dst)
        k_edge_agg<<<nblk((long long)N_EDGES * 32, 256), 256, 0, stream>>>(ei, deg, t, agg);
        // batch-norm + relu -> h
        k_fill<<<1, 256, 0, stream>>>(colsum, 0.0f, H);
        k_fill<<<1, 256, 0, stream>>>(colsq, 0.0f, H);
        k_bn_reduce<<<nblk(N_NODES, 64), 256, 0, stream>>>(agg, colsum, colsq);
        k_bn_apply <<<nblk((long long)N_NODES * H, 256), 256, 0, stream>>>(
            agg, colsum, colsq, bnG + i * H, bnB + i * H, h);
    }

    // global mean pool -> gsum[G,H]
    k_fill<<<nblk(GNUM * H + GNUM, 256), 256, 0, stream>>>(gsum, 0.0f, GNUM * H + GNUM);
    k_cnt<<<nblk(N_NODES, 256), 256, 0, stream>>>(batch, gcnt);
    k_pool_accum<<<nblk((long long)N_NODES * H, 256), 256, 0, stream>>>(h, batch, gsum);
    k_pool_div<<<nblk(GNUM * H, 256), 256, 0, stream>>>(gsum, gcnt);

    // fc head
    k_gemm256<<<GNUM / 32, 256, 0, stream>>>(gsum, fcW1, fcB1, m1, H);
    k_bn_small<<<1, H, 0, stream>>>(m1, fcG1, fcBe1, m1, 1);
    k_gemm256<<<GNUM / 32, 256, 0, stream>>>(m1, fcW2, fcB2, m2, H);
    k_bn_small<<<1, H, 0, stream>>>(m2, fcG2, fcBe2, m2, 0);
    k_fc3<<<nblk(GNUM * C_OUT, 256), 256, 0, stream>>>(m2, fcW3, fcB3, (float*)d_out);
}